// NeuraLogicLayer_64750926954840
// MI455X (gfx1250) — compile-verified
//
#include <hip/hip_runtime.h>

#define DDIM 16
#define NW   256

typedef __attribute__((ext_vector_type(2))) float v2f;
typedef __attribute__((ext_vector_type(8))) float v8f;

// ---------------------------------------------------------------------------
// 1) LDS-privatized histogram of widx
// ---------------------------------------------------------------------------
__global__ void hist_kernel(const int* __restrict__ widx, int* __restrict__ counts, int E) {
    __shared__ int sh[NW];
    for (int i = threadIdx.x; i < NW; i += blockDim.x) sh[i] = 0;
    __syncthreads();
    int i = blockIdx.x * blockDim.x + threadIdx.x;
    int stride = gridDim.x * blockDim.x;
    for (; i < E; i += stride) atomicAdd(&sh[widx[i] & (NW - 1)], 1);
    __syncthreads();
    for (int i2 = threadIdx.x; i2 < NW; i2 += blockDim.x)
        if (sh[i2]) atomicAdd(&counts[i2], sh[i2]);
}

// ---------------------------------------------------------------------------
// 2) Exclusive scan over 256 bins (trivial, runs once)
// ---------------------------------------------------------------------------
__global__ void scan_kernel(const int* __restrict__ counts,
                            int* __restrict__ offsets, int* __restrict__ cursor) {
    if (threadIdx.x == 0) {
        int acc = 0;
        for (int i = 0; i < NW; ++i) { offsets[i] = acc; cursor[i] = acc; acc += counts[i]; }
        offsets[NW] = acc;
    }
}

// ---------------------------------------------------------------------------
// 3) Block-aggregated scatter of edge IDs into per-weight bins
// ---------------------------------------------------------------------------
__global__ void scatter_kernel(const int* __restrict__ widx, int* __restrict__ cursor,
                               int* __restrict__ bins, int E) {
    __shared__ int sh_cnt[NW];
    __shared__ int sh_base[NW];
    int chunk = (E + gridDim.x - 1) / gridDim.x;
    int lo = blockIdx.x * chunk;
    int hi = min(lo + chunk, E);

    for (int i = threadIdx.x; i < NW; i += blockDim.x) sh_cnt[i] = 0;
    __syncthreads();
    for (int e = lo + threadIdx.x; e < hi; e += blockDim.x)
        atomicAdd(&sh_cnt[widx[e] & (NW - 1)], 1);
    __syncthreads();
    for (int i = threadIdx.x; i < NW; i += blockDim.x) {
        int c = sh_cnt[i];
        sh_base[i] = c ? atomicAdd(&cursor[i], c) : 0;
        sh_cnt[i] = 0;   // reuse as local cursor
    }
    __syncthreads();
    for (int e = lo + threadIdx.x; e < hi; e += blockDim.x) {
        int w = widx[e] & (NW - 1);
        int pos = sh_base[w] + atomicAdd(&sh_cnt[w], 1);
        bins[pos] = e;
    }
}

// ---------------------------------------------------------------------------
// 4) WMMA edge-transform + atomic scatter.
//    One wave per (weight bin, y-slice). Tile = 16 edges sharing W[bin]:
//      msgs(16x16) = W(16x16) @ X(16x16)  via 4x V_WMMA_F32_16X16X4_F32
//    Out-of-range columns are computed on clamped (garbage) data and simply
//    not scattered: D[:,n] depends only on B[:,n], so no masking is needed.
//    Index chain (bins -> u/v) is software-pipelined one tile
//    ahead; in-order LOADcnt completion means the next tile's B-operand wait
//    subsumes the index-chain loads, so no explicit wait/pin is needed.
// ---------------------------------------------------------------------------
__global__ __launch_bounds__(32) void edge_wmma_kernel(
    const float* __restrict__ x, const float* __restrict__ W,
    const int* __restrict__ u, const int* __restrict__ v,
    const int* __restrict__ bins, const int* __restrict__ offsets,
    float* __restrict__ out)
{
    const int w     = blockIdx.x;          // weight index 0..255
    const int start = offsets[w];
    const int end   = offsets[w + 1];
    const int lane  = threadIdx.x;         // 0..31 (wave32)
    const int col   = lane & 15;           // matrix column / row owned by this lane
    const int koff  = (lane >> 4) << 1;    // 0 for lanes 0-15, 2 for lanes 16-31

    const int step = (int)gridDim.y << 4;
    int t = start + ((int)blockIdx.y << 4);
    if (t >= end) return;

    // A-matrix (W) in ISA 16x4 f32 layout, hoisted across all tiles of this bin:
    // lane<16 : row=lane,   VGPR0=W[row][4s+0], VGPR1=W[row][4s+1]
    // lane>=16: row=lane-16, VGPR0=W[row][4s+2], VGPR1=W[row][4s+3]
    const float* Wp = W + (size_t)w * (DDIM * DDIM);
    v2f a[4];
#pragma unroll
    for (int s = 0; s < 4; ++s)
        a[s] = *(const v2f*)(Wp + col * DDIM + 4 * s + koff);

    // --- pipeline prologue: resolve first tile's full index chain ---
    int idx  = t + col;
    int sIdx = idx < end ? idx : end - 1;       // clamped, always in-bin
    int eid  = bins[sIdx];
    int src  = u[eid];
    int dst  = v[eid];

    for (;;) {
        const bool valid = (t + col) < end;

        // Unconditional b64 gathers of this lane's half-row of x[src].
        const float* xr = x + (size_t)src * DDIM + koff;
        v2f b0 = *(const v2f*)(xr + 0);
        v2f b1 = *(const v2f*)(xr + 4);
        v2f b2 = *(const v2f*)(xr + 8);
        v2f b3 = *(const v2f*)(xr + 12);

        // Issue next tile's index chain (bins -> u,v) so it overlaps the WMMAs;
        // its completion is absorbed by the next iteration's B-operand wait.
        const int  tn   = t + step;
        const bool more = tn < end;
        int nIdx  = tn + col;
        int nsIdx = more ? (nIdx < end ? nIdx : end - 1) : sIdx;
        int neid  = bins[nsIdx];
        int nsrc  = u[neid];
        int ndst  = v[neid];

        v8f c = {};
        c = __builtin_amdgcn_wmma_f32_16x16x4_f32(false, a[0], false, b0, (short)0, c, false, false);
        c = __builtin_amdgcn_wmma_f32_16x16x4_f32(false, a[1], false, b1, (short)0, c, false, false);
        c = __builtin_amdgcn_wmma_f32_16x16x4_f32(false, a[2], false, b2, (short)0, c, false, false);
        c = __builtin_amdgcn_wmma_f32_16x16x4_f32(false, a[3], false, b3, (short)0, c, false, false);

        // D layout: VGPR j -> lanes 0-15 hold (M=j, N=lane); lanes 16-31 (M=j+8).
        // Lane owns output rows base..base+7 of column `col` -> 8 atomic adds.
        if (valid) {
            float* op = out + (size_t)dst * DDIM + ((lane >> 4) << 3);
#pragma unroll
            for (int j = 0; j < 8; ++j)
                atomicAdd(op + j, c[j]);
        }

        if (!more) break;
        t = tn; sIdx = nsIdx; eid = neid; src = nsrc; dst = ndst;
    }
}

// ---------------------------------------------------------------------------
// 5) ReLU epilogue
// ---------------------------------------------------------------------------
__global__ void relu_kernel(float* __restrict__ out, int n) {
    int i = blockIdx.x * blockDim.x + threadIdx.x;
    if (i < n) out[i] = fmaxf(out[i], 0.0f);
}

// ---------------------------------------------------------------------------
// Fallback (only if workspace is too small to bin edges): naive per-edge matvec
// ---------------------------------------------------------------------------
__global__ void edge_naive_kernel(
    const float* __restrict__ x, const float* __restrict__ W,
    const int* __restrict__ u, const int* __restrict__ v,
    const int* __restrict__ widx, float* __restrict__ out, int E)
{
    int e = blockIdx.x * blockDim.x + threadIdx.x;
    if (e >= E) return;
    const float* xr = x + (size_t)u[e] * DDIM;
    const float* Wp = W + (size_t)(widx[e] & (NW - 1)) * (DDIM * DDIM);
    float xi[DDIM];
#pragma unroll
    for (int i = 0; i < DDIM; ++i) xi[i] = xr[i];
    float* op = out + (size_t)v[e] * DDIM;
#pragma unroll
    for (int o = 0; o < DDIM; ++o) {
        float acc = 0.0f;
#pragma unroll
        for (int i = 0; i < DDIM; ++i) acc += Wp[o * DDIM + i] * xi[i];
        atomicAdd(op + o, acc);
    }
}

// ---------------------------------------------------------------------------
extern "C" void kernel_launch(void* const* d_in, const int* in_sizes, int n_in,
                              void* d_out, int out_size, void* d_ws, size_t ws_size,
                              hipStream_t stream) {
    const float* x    = (const float*)d_in[0];
    const float* W    = (const float*)d_in[1];
    const int*   u    = (const int*)d_in[2];
    const int*   v    = (const int*)d_in[3];
    const int*   widx = (const int*)d_in[4];
    float* out = (float*)d_out;
    const int E = in_sizes[2];

    // out accumulates via atomics -> must start at zero every call
    hipMemsetAsync(d_out, 0, (size_t)out_size * sizeof(float), stream);

    const size_t meta_ints = NW /*counts*/ + (NW + 1) /*offsets*/ + NW /*cursor*/;
    const size_t need = (meta_ints + (size_t)E) * sizeof(int);

    if (ws_size >= need) {
        int* counts  = (int*)d_ws;
        int* offsets = counts + NW;
        int* cursor  = offsets + NW + 1;
        int* bins    = cursor + NW;

        hipMemsetAsync(counts, 0, NW * sizeof(int), stream);

        hist_kernel<<<1024, 256, 0, stream>>>(widx, counts, E);
        scan_kernel<<<1, 32, 0, stream>>>(counts, offsets, cursor);
        scatter_kernel<<<1024, 256, 0, stream>>>(widx, cursor, bins, E);

        dim3 grid(NW, 32);   // 256 bins x 32 slices = 8192 waves
        edge_wmma_kernel<<<grid, 32, 0, stream>>>(x, W, u, v, bins, offsets, out);
    } else {
        edge_naive_kernel<<<(E + 255) / 256, 256, 0, stream>>>(x, W, u, v, widx, out, E);
    }

    relu_kernel<<<(out_size + 255) / 256, 256, 0, stream>>>(out, out_size);
}